// ProtoCycleModel_60627758350549
// MI455X (gfx1250) — compile-verified
//
#include <hip/hip_runtime.h>
#include <hip/hip_bf16.h>

// ProtoCycleModel on MI455X (gfx1250): fp32 WMMA (V_WMMA_F32_16X16X4_F32) with
// double-buffered GLOBAL_LOAD_ASYNC_TO_LDS_B128 staging (ASYNCcnt pipeline).
// P=65536 rows, C=64 classes, D=256, n=1024 rows per class.

typedef __attribute__((ext_vector_type(2))) float v2f;
typedef __attribute__((ext_vector_type(8))) float v8f;

#define NP 65536
#define NC 64
#define ND 256
#define NPC 1024        // rows per class
#define BK 16           // k-chunk staged in LDS per pipeline stage
#define NKC (ND / BK)   // 16 k-chunks
#define LDK 20          // padded LDS row stride (floats): conflict-free frag reads

// ---- gfx1250 async global->LDS copy (tracked by ASYNCcnt) -------------------
// VDST = LDS byte offset (low 32 bits of the flat LDS address), VADDR = 64-bit VA.
__device__ __forceinline__ void async_b128(void* lds, const void* glob) {
    unsigned loff = (unsigned)(uintptr_t)lds;  // LDS aperture: addr[31:0] = LDS offset
    asm volatile("global_load_async_to_lds_b128 %0, %1, off"
                 :: "v"(loff), "v"(glob) : "memory");
}

#if __has_builtin(__builtin_amdgcn_s_wait_asynccnt)
#define WAIT_ASYNC(n) __builtin_amdgcn_s_wait_asynccnt(n)
#else
#define WAIT_ASYNC(n) asm volatile("s_wait_asynccnt " #n ::: "memory")
#endif

// ---------------------------------------------------------------- zero output
__global__ void zero_out_kernel(float* __restrict__ out) {
    if (threadIdx.x < 2 * NC) out[threadIdx.x] = 0.0f;
}

// ------------------------------------------------- 256x256 inverse (Gauss-Jordan)
// One block of 1024 threads; matrix in registers (64/thread), pivot row/column
// broadcast via 2KB LDS. No pivoting (W is dense Gaussian, a.s. invertible).
__global__ __launch_bounds__(1024) void invert256_kernel(const float* __restrict__ W,
                                                         float* __restrict__ invW) {
    __shared__ float colk[256];
    __shared__ float rowk[256];
    const int t = threadIdx.x;
    const int j = t & 255;
    const int row0 = (t >> 8) * 64;

    float A[64];
#pragma unroll
    for (int i = 0; i < 64; ++i) A[i] = W[(row0 + i) * ND + j];

    for (int k = 0; k < 256; ++k) {
        if (j == k) {
#pragma unroll
            for (int i = 0; i < 64; ++i) colk[row0 + i] = A[i];
        }
        __syncthreads();
        const float inv = 1.0f / colk[k];
        if (k >= row0 && k < row0 + 64) {
#pragma unroll
            for (int i = 0; i < 64; ++i) {
                if (row0 + i == k) {
                    const float v = (j == k) ? inv : A[i] * inv;
                    A[i] = v;
                    rowk[j] = v;
                }
            }
        }
        __syncthreads();
        const float rkj = rowk[j];
#pragma unroll
        for (int i = 0; i < 64; ++i) {
            const int gi = row0 + i;
            if (gi != k) {
                const float base = (j == k) ? 0.0f : A[i];
                A[i] = base - colk[gi] * rkj;
            }
        }
        __syncthreads();
    }
#pragma unroll
    for (int i = 0; i < 64; ++i) invW[(row0 + i) * ND + j] = A[i];
}

// ------------------------------------- cvec[d] = -sum_e b[e]*invW[d,e]  (fold bias)
__global__ __launch_bounds__(256) void bias_fold_kernel(const float* __restrict__ invW,
                                                        const float* __restrict__ b,
                                                        float* __restrict__ cvec) {
    const int d = threadIdx.x;
    float s = 0.0f;
    for (int e = 0; e < ND; ++e) s += b[e] * invW[d * ND + e];
    cvec[d] = -s;
}

// ------------------------------------------------------------- row squared norms
__global__ __launch_bounds__(256) void rownorm_kernel(const float* __restrict__ X,
                                                      float* __restrict__ norms) {
    const int lane = threadIdx.x & 31;
    const int wave = threadIdx.x >> 5;
    const long row = (long)blockIdx.x * 8 + wave;
    const float* p = X + row * ND;
    float s = 0.0f;
#pragma unroll
    for (int q = 0; q < 8; ++q) {
        const float v = p[lane + q * 32];
        s += v * v;
    }
    s += __shfl_xor(s, 16, 32);
    s += __shfl_xor(s, 8, 32);
    s += __shfl_xor(s, 4, 32);
    s += __shfl_xor(s, 2, 32);
    s += __shfl_xor(s, 1, 32);
    if (lane == 0) norms[row] = s;
}

// ----------------------------------------------------- T = X @ M^T + cvec
// 128 rows/block, 2 j-tiles of 128; async double-buffered LDS staging; fused
// |T row|^2 epilogue. Wave w owns rows w*16..w*16+15, 8 column frags per tile.
__global__ __launch_bounds__(256) void transform_kernel(const float* __restrict__ X,
                                                        const float* __restrict__ M,
                                                        const float* __restrict__ cvec,
                                                        float* __restrict__ T,
                                                        float* __restrict__ rnorm) {
    __shared__ float As[2][128][LDK];
    __shared__ float Bs[2][128][LDK];
    const int t = threadIdx.x;
    const int lane = t & 31;
    const int wave = t >> 5;
    const int hi = lane >> 4;
    const int n = lane & 15;
    const int rbase = wave * 16;
    const long rowBlock = (long)blockIdx.x * 128;

    float rnsum[8];
#pragma unroll
    for (int v = 0; v < 8; ++v) rnsum[v] = 0.0f;

    for (int jt = 0; jt < 2; ++jt) {
        const int j0 = jt * 128;
        v8f acc[8];
#pragma unroll
        for (int jj = 0; jj < 8; ++jj) { v8f z = {}; acc[jj] = z; }

        // prologue: stage 0 into buffer 0
#pragma unroll
        for (int q = 0; q < 2; ++q) {
            const int idx = t + q * 256;
            const int r = idx >> 2;
            const int s = (idx & 3) * 4;
            async_b128(&As[0][r][s], X + (rowBlock + r) * ND + s);
            async_b128(&Bs[0][r][s], M + (long)(j0 + r) * ND + s);
        }

        for (int kc = 0; kc < NKC; ++kc) {
            const int cur = kc & 1;
            if (kc + 1 < NKC) {
                const int k0n = (kc + 1) * BK;
#pragma unroll
                for (int q = 0; q < 2; ++q) {
                    const int idx = t + q * 256;
                    const int r = idx >> 2;
                    const int s = (idx & 3) * 4;
                    async_b128(&As[cur ^ 1][r][s], X + (rowBlock + r) * ND + k0n + s);
                    async_b128(&Bs[cur ^ 1][r][s], M + (long)(j0 + r) * ND + k0n + s);
                }
                WAIT_ASYNC(4);   // stage kc complete (in-order), kc+1 in flight
            } else {
                WAIT_ASYNC(0);
            }
            __syncthreads();
#pragma unroll
            for (int ks = 0; ks < BK / 4; ++ks) {
                const int kk = ks * 4 + 2 * hi;
                v2f a;
                a.x = As[cur][rbase + n][kk];
                a.y = As[cur][rbase + n][kk + 1];
#pragma unroll
                for (int jj = 0; jj < 8; ++jj) {
                    v2f bfr;
                    bfr.x = Bs[cur][jj * 16 + n][kk];
                    bfr.y = Bs[cur][jj * 16 + n][kk + 1];
                    acc[jj] = __builtin_amdgcn_wmma_f32_16x16x4_f32(
                        false, a, false, bfr, (short)0, acc[jj], false, false);
                }
            }
            __syncthreads();   // all reads done before buffer reuse at kc+2
        }
        // epilogue: write T tile + accumulate row-norm partials
#pragma unroll
        for (int jj = 0; jj < 8; ++jj) {
            const int col = j0 + jj * 16 + n;
            const float badd = cvec[col];
#pragma unroll
            for (int v = 0; v < 8; ++v) {
                const long grow = rowBlock + rbase + v + 8 * hi;
                const float val = acc[jj][v] + badd;
                T[grow * ND + col] = val;
                rnsum[v] += val * val;
            }
        }
    }
#pragma unroll
    for (int v = 0; v < 8; ++v) {
        float s = rnsum[v];
        s += __shfl_xor(s, 1, 32);
        s += __shfl_xor(s, 2, 32);
        s += __shfl_xor(s, 4, 32);
        s += __shfl_xor(s, 8, 32);
        if (n == 0) {
            const long grow = rowBlock + rbase + v + 8 * hi;
            rnorm[grow] = s;
        }
    }
}

// -------------------------- per-class fused GEMM + rowmin + mean
// Block = (class c, 128-row i-block); 8 j-tiles of 128 over the class's 1024
// candidates; running min_j(|y_j|^2 - 2 x.y) kept in C-fragment layout.
__global__ __launch_bounds__(256) void classmin_kernel(const float* __restrict__ Xsrc,
                                                       const float* __restrict__ Tm,
                                                       const float* __restrict__ normX,
                                                       const float* __restrict__ normY,
                                                       float* __restrict__ out) {
    __shared__ float As[2][128][LDK];
    __shared__ float Bs[2][128][LDK];
    __shared__ float red[16];
    const int t = threadIdx.x;
    const int lane = t & 31;
    const int wave = t >> 5;
    const int hi = lane >> 4;
    const int n = lane & 15;
    const int rbase = wave * 16;
    const int i0 = blockIdx.x * 128;   // row block within class
    const int c = blockIdx.y;          // class id

    float rmin[8];
#pragma unroll
    for (int v = 0; v < 8; ++v) rmin[v] = 3.0e38f;

    for (int jt = 0; jt < 8; ++jt) {
        const int j0 = jt * 128;
        v8f acc[8];
#pragma unroll
        for (int jj = 0; jj < 8; ++jj) { v8f z = {}; acc[jj] = z; }

        // prologue: stage 0 into buffer 0 (class-strided row gather)
#pragma unroll
        for (int q = 0; q < 2; ++q) {
            const int idx = t + q * 256;
            const int r = idx >> 2;
            const int s = (idx & 3) * 4;
            async_b128(&As[0][r][s], Xsrc + ((long)(i0 + r) * NC + c) * ND + s);
            async_b128(&Bs[0][r][s], Tm + ((long)(j0 + r) * NC + c) * ND + s);
        }

        for (int kc = 0; kc < NKC; ++kc) {
            const int cur = kc & 1;
            if (kc + 1 < NKC) {
                const int k0n = (kc + 1) * BK;
#pragma unroll
                for (int q = 0; q < 2; ++q) {
                    const int idx = t + q * 256;
                    const int r = idx >> 2;
                    const int s = (idx & 3) * 4;
                    async_b128(&As[cur ^ 1][r][s],
                               Xsrc + ((long)(i0 + r) * NC + c) * ND + k0n + s);
                    async_b128(&Bs[cur ^ 1][r][s],
                               Tm + ((long)(j0 + r) * NC + c) * ND + k0n + s);
                }
                WAIT_ASYNC(4);
            } else {
                WAIT_ASYNC(0);
            }
            __syncthreads();
#pragma unroll
            for (int ks = 0; ks < BK / 4; ++ks) {
                const int kk = ks * 4 + 2 * hi;
                v2f a;
                a.x = As[cur][rbase + n][kk];
                a.y = As[cur][rbase + n][kk + 1];
#pragma unroll
                for (int jj = 0; jj < 8; ++jj) {
                    v2f bfr;
                    bfr.x = Bs[cur][jj * 16 + n][kk];
                    bfr.y = Bs[cur][jj * 16 + n][kk + 1];
                    acc[jj] = __builtin_amdgcn_wmma_f32_16x16x4_f32(
                        false, a, false, bfr, (short)0, acc[jj], false, false);
                }
            }
            __syncthreads();
        }
        // min epilogue for this j-tile (defer +|x_i|^2 to the end)
#pragma unroll
        for (int jj = 0; jj < 8; ++jj) {
            const int jl = j0 + jj * 16 + n;
            const float ny = normY[(long)jl * NC + c];
#pragma unroll
            for (int v = 0; v < 8; ++v) {
                const float d = ny - 2.0f * acc[jj][v];
                rmin[v] = fminf(rmin[v], d);
            }
        }
    }
    // cross-lane min over the 16 N-lanes of each half, then block mean
    float rowsum = 0.0f;
#pragma unroll
    for (int v = 0; v < 8; ++v) {
        float m = rmin[v];
        m = fminf(m, __shfl_xor(m, 1, 32));
        m = fminf(m, __shfl_xor(m, 2, 32));
        m = fminf(m, __shfl_xor(m, 4, 32));
        m = fminf(m, __shfl_xor(m, 8, 32));
        if (n == 0) {
            const long grow = (long)(i0 + rbase + v + 8 * hi) * NC + c;
            rowsum += m + normX[grow];
        }
    }
    if (n == 0) red[wave * 2 + hi] = rowsum;
    __syncthreads();
    if (t == 0) {
        float s = 0.0f;
#pragma unroll
        for (int u = 0; u < 16; ++u) s += red[u];
        atomicAdd(out + c, s * (1.0f / (float)NPC));
    }
}

// ------------------------------------------------------------------ launcher
extern "C" void kernel_launch(void* const* d_in, const int* in_sizes, int n_in,
                              void* d_out, int out_size, void* d_ws, size_t ws_size,
                              hipStream_t stream) {
    const float* protos1 = (const float*)d_in[0];
    const float* protos2 = (const float*)d_in[1];
    const float* W       = (const float*)d_in[2];
    const float* b       = (const float*)d_in[3];
    float* out = (float*)d_out;

    // ws layout (floats): invW[65536] | normX[65536] | normY[65536] | cvec[256] | T[16777216]
    float* ws    = (float*)d_ws;
    float* invW  = ws;
    float* normX = ws + 65536;
    float* normY = ws + 131072;
    float* cvecA = ws + 196608;
    float* T     = ws + 196864;

    zero_out_kernel<<<1, 128, 0, stream>>>(out);
    invert256_kernel<<<1, 1024, 0, stream>>>(W, invW);
    bias_fold_kernel<<<1, 256, 0, stream>>>(invW, b, cvecA);

    // pass A: a = dist(p1, (p2 - b) @ inv(W)^T) = dist(p1, p2@invW^T + cvecA)
    rownorm_kernel<<<NP / 8, 256, 0, stream>>>(protos1, normX);
    transform_kernel<<<NP / 128, 256, 0, stream>>>(protos2, invW, cvecA, T, normY);
    classmin_kernel<<<dim3(NPC / 128, NC), 256, 0, stream>>>(protos1, T, normX, normY,
                                                             out + 0);

    // pass B: bb = dist(p2, p1 @ W^T + b)
    rownorm_kernel<<<NP / 8, 256, 0, stream>>>(protos2, normX);
    transform_kernel<<<NP / 128, 256, 0, stream>>>(protos1, W, b, T, normY);
    classmin_kernel<<<dim3(NPC / 128, NC), 256, 0, stream>>>(protos2, T, normX, normY,
                                                             out + NC);
}